// PhasorDiscriminator_59554016526921
// MI455X (gfx1250) — compile-verified
//
#include <hip/hip_runtime.h>
#include <math.h>

typedef __attribute__((ext_vector_type(2))) float v2f;
typedef __attribute__((ext_vector_type(8))) float v8f;

namespace {
constexpr int TN = 1024;            // sequence length / DFT size
constexpr int LN = 6;               // layers
constexpr int SPW = 8;              // samples per workgroup (1 per wave)
constexpr int THREADS = 256;        // 8 waves (wave32)
}

// D = A(16x4) * B(4x16) + C, fp32 WMMA (CDNA5)
__device__ __forceinline__ v8f wmma_f32(v2f a, v2f b, v8f c) {
    return __builtin_amdgcn_wmma_f32_16x16x4_f32(
        /*neg_a=*/false, a, /*neg_b=*/false, b,
        /*c_mod=*/(short)0, c, /*reuse_a=*/false, /*reuse_b=*/false);
}

// z state lives in a "VT" plane layout: linear index n = n2 + 32*n1  ->  plane[n2*32 + n1].
// Every WMMA A-fragment (two K-adjacent elements of one row) is one aligned b64 load.

__global__ __launch_bounds__(THREADS)
void phasor_disc_kernel(const float* __restrict__ X,
                        const float* __restrict__ W,
                        float* __restrict__ out)
{
    // ---- dynamic LDS carve-up, SoA planes (total 192 KB) ----
    extern __shared__ float smem[];
    float*  sDre  = smem;                       // 32x32 Re(DFT32), row-major [m][k]
    float*  sDim  = sDre + 1024;                // 32x32 Im(DFT32)
    float4* sTW4  = (float4*)(sDim + 1024);     // [k1][n2pair]: (c0,c1,s0,s1)/32, 512 entries
    float*  sWre  = (float*)(sTW4 + 512);       // 6*1024 Re(exp(iw)/sqrt2), VT-permuted
    float*  sWim  = sWre + LN * TN;             // 6*1024 Im(exp(iw)/sqrt2), VT-permuted
    float*  sZre  = sWim + LN * TN;             // 8 x 1024 Re(z), VT layout
    float*  sZim  = sZre + SPW * TN;            // 8 x 1024 Im(z), VT layout
    float*  sBre  = sZim + SPW * TN;            // 8 x (32x32) Re(B), [k1][n2]
    float*  sBim  = sBre + SPW * 1024;          // 8 x (32x32) Im(B), [k1][n2]

    const int tid  = threadIdx.x;
    const int lane = tid & 31;
    const int wave = tid >> 5;
    const int ml   = lane & 15;     // M/N index within a 16-wide tile
    const int half = lane >> 4;     // upper half: K+2 for A/B frags, M+8 for C/D
    const int koff = half * 2;

    const float PI2 = 6.2831853071795864769f;
    const float inv_sqrt2 = 0.70710678118654752440f;

    // ---- one-time table setup (cooperative, then barrier) ----
    for (int i = tid; i < 1024; i += THREADS) {
        int m = i >> 5, k = i & 31;
        float ang = -PI2 * (float)((m * k) & 31) * (1.0f / 32.0f);
        float sn, cs; __sincosf(ang, &sn, &cs);
        sDre[i] = cs; sDim[i] = sn;
    }
    // twiddle LUT: entry [k1*16 + n2/2] = (cos t0, cos t1, sin t0, sin t1)/32,
    // t0 = k1*n2, t1 = k1*(n2+1), with 1/sqrt(1024) folded in
    for (int i = tid; i < 512; i += THREADS) {
        int k1 = i >> 4, c2 = i & 15;
        int t0 = (k1 * (2 * c2)) & 1023;
        int t1 = (k1 * (2 * c2 + 1)) & 1023;
        float s0, c0, s1, c1;
        __sincosf(-PI2 * (float)t0 * (1.0f / 1024.0f), &s0, &c0);
        __sincosf(-PI2 * (float)t1 * (1.0f / 1024.0f), &s1, &c1);
        sTW4[i] = make_float4(c0 * (1.0f / 32.0f), c1 * (1.0f / 32.0f),
                              s0 * (1.0f / 32.0f), s1 * (1.0f / 32.0f));
    }
    // exp(i*w)/sqrt2, VT-permuted (folds the mixer's 1/sqrt2 into the shift)
    for (int i = tid; i < LN * TN; i += THREADS) {
        int l = i >> 10, n = i & 1023;
        int a = l * TN + (n & 31) * 32 + (n >> 5);
        float w = W[i];
        float sn, cs; __sincosf(w, &sn, &cs);
        sWre[a] = cs * inv_sqrt2; sWim[a] = sn * inv_sqrt2;
    }

    // ---- per-wave sample ownership ----
    const int s = blockIdx.x * SPW + wave;
    float* zre  = sZre + wave * TN;
    float* zim  = sZim + wave * TN;
    float* btre = sBre + wave * 1024;
    float* btim = sBim + wave * 1024;
    const float* xrow = X + (size_t)s * TN;

    __builtin_prefetch(xrow + lane * 4, 0, 3);  // global_prefetch_b8

    // encode: z = exp(i*arcsin(x)) = (sqrt(1-x^2), x), x clipped; scatter into VT layout
    #pragma unroll 4
    for (int it = 0; it < TN / 32; ++it) {
        int n = lane + it * 32;                 // n>>5 == it
        float x = xrow[n];
        x = fminf(0.999f, fmaxf(-0.999f, x));
        int a = (n & 31) * 32 + it;
        zre[a] = sqrtf(fmaxf(0.0f, 1.0f - x * x));
        zim[a] = x;
    }
    __syncthreads();   // tables visible to all waves; z is wave-private after this

    // ---- preload constant DFT-32 B-fragments (Re/Im: 64 VGPRs, layer-invariant) ----
    v2f dRe[2][8], dIm[2][8];
    #pragma unroll
    for (int nt = 0; nt < 2; ++nt) {
        const int ncol = nt * 16 + ml;
        #pragma unroll
        for (int kc = 0; kc < 8; ++kc) {
            int kb = kc * 4 + koff;
            v2f re; re.x = sDre[kb * 32 + ncol]; re.y = sDre[(kb + 1) * 32 + ncol];
            v2f im; im.x = sDim[kb * 32 + ncol]; im.y = sDim[(kb + 1) * 32 + ncol];
            dRe[nt][kc] = re; dIm[nt][kc] = im;
        }
    }

    #pragma unroll 1
    for (int l = 0; l < LN; ++l) {
        const float* wre = sWre + l * TN;
        const float* wim = sWim + l * TN;

        // ---- shift by exp(i*w)/sqrt2 then pair mix, in VT layout; packed fp32 math.
        //      Pair (2p,2p+1), 2p = 2c + 32d: plane[2c*32+d] & plane[(2c+1)*32+d].
        //      Two pairs (d, d+1) per iteration -> all accesses are aligned b64. ----
        #pragma unroll 2
        for (int it = 0; it < 8; ++it) {
            int c = (lane >> 4) + 2 * it;       // n2-pair index, 0..15
            int d = (lane & 15) * 2;            // n1 base (even)
            int e = c * 64 + d;                 // plane addr of (n2=2c, n1=d)
            v2f r0  = *(v2f*)(zre + e);
            v2f r1  = *(v2f*)(zre + e + 32);
            v2f i0  = *(v2f*)(zim + e);
            v2f i1  = *(v2f*)(zim + e + 32);
            v2f wr0 = *(v2f*)(wre + e);
            v2f wr1 = *(v2f*)(wre + e + 32);
            v2f wi0 = *(v2f*)(wim + e);
            v2f wi1 = *(v2f*)(wim + e + 32);
            v2f ar = r0 * wr0 - i0 * wi0;       // a = z*exp(iw)/sqrt2  (v_pk_* ops)
            v2f ai = r0 * wi0 + i0 * wr0;
            v2f br = r1 * wr1 - i1 * wi1;       // b = partner*exp(iw)/sqrt2
            v2f bi = r1 * wi1 + i1 * wr1;
            *(v2f*)(zre + e)      = ar + br;
            *(v2f*)(zre + e + 32) = ar - br;
            *(v2f*)(zim + e)      = ai + bi;
            *(v2f*)(zim + e + 32) = ai - bi;
        }

        // ---- phase 1: out1 = V^T * D  (== (D*V)^T, D symmetric), data in A operand.
        //      out1[n2][k1], twiddled (packed, folded /32), stored to bt[k1][n2]. ----
        #pragma unroll
        for (int mt = 0; mt < 2; ++mt)
        #pragma unroll
        for (int nt = 0; nt < 2; ++nt) {
            v8f accRe = {}; v8f accIm = {};
            #pragma unroll
            for (int kc = 0; kc < 8; ++kc) {
                const int abase = (mt * 16 + ml) * 32 + kc * 4 + koff;
                v2f aRe  = *(const v2f*)(zre + abase);
                v2f aIm  = *(const v2f*)(zim + abase);
                v2f aImN = -aIm;                                 // for -Im*Im
                accRe = wmma_f32(aRe,  dRe[nt][kc], accRe);      //  Re*Re
                accIm = wmma_f32(aIm,  dRe[nt][kc], accIm);      //  Im*Re
                accRe = wmma_f32(aImN, dIm[nt][kc], accRe);      // -Im*Im
                accIm = wmma_f32(aRe,  dIm[nt][kc], accIm);      //  Re*Im
            }
            const int k1   = nt * 16 + ml;
            const int base = k1 * 32 + mt * 16 + half * 8;       // bt[k1][n2..n2+7]
            float tre[8], tim[8];
            #pragma unroll
            for (int j = 0; j < 4; ++j) {                        // r-pairs (2j, 2j+1)
                float4 tw = sTW4[k1 * 16 + mt * 8 + half * 4 + j];
                v2f ar; ar.x = accRe[2 * j]; ar.y = accRe[2 * j + 1];
                v2f ai; ai.x = accIm[2 * j]; ai.y = accIm[2 * j + 1];
                v2f cc; cc.x = tw.x; cc.y = tw.y;
                v2f ss; ss.x = tw.z; ss.y = tw.w;
                v2f tr = ar * cc - ai * ss;                      // packed
                v2f ti = ar * ss + ai * cc;
                tre[2 * j] = tr.x; tre[2 * j + 1] = tr.y;
                tim[2 * j] = ti.x; tim[2 * j + 1] = ti.y;
            }
            *(float4*)(btre + base)     = make_float4(tre[0], tre[1], tre[2], tre[3]);
            *(float4*)(btre + base + 4) = make_float4(tre[4], tre[5], tre[6], tre[7]);
            *(float4*)(btim + base)     = make_float4(tim[0], tim[1], tim[2], tim[3]);
            *(float4*)(btim + base + 4) = make_float4(tim[4], tim[5], tim[6], tim[7]);
        }

        // ---- phase 2: out2 = B * D, data in A operand from bt[k1][n2].
        //      out2[k1][k2] -> z linear [k1 + 32*k2] == VT plane [k1*32 + k2]. ----
        #pragma unroll
        for (int mt = 0; mt < 2; ++mt)
        #pragma unroll
        for (int nt = 0; nt < 2; ++nt) {
            v8f accRe = {}; v8f accIm = {};
            #pragma unroll
            for (int kc = 0; kc < 8; ++kc) {
                const int abase = (mt * 16 + ml) * 32 + kc * 4 + koff;
                v2f aRe  = *(const v2f*)(btre + abase);
                v2f aIm  = *(const v2f*)(btim + abase);
                v2f aImN = -aIm;
                accRe = wmma_f32(aRe,  dRe[nt][kc], accRe);
                accIm = wmma_f32(aIm,  dRe[nt][kc], accIm);
                accRe = wmma_f32(aImN, dIm[nt][kc], accRe);
                accIm = wmma_f32(aRe,  dIm[nt][kc], accIm);
            }
            #pragma unroll
            for (int r = 0; r < 8; ++r) {
                int a = (mt * 16 + r + half * 8) * 32 + nt * 16 + ml;
                zre[a] = accRe[r];
                zim[a] = accIm[r];
            }
        }
    }

    // ---- readout: linear z[0] == VT plane[0]; (sin(angle) + 1)/2 = (Im/|z| + 1)/2 ----
    if (lane == 0) {
        float re0 = zre[0], im0 = zim[0];
        float mag = sqrtf(re0 * re0 + im0 * im0) + 1e-30f;
        out[s] = (im0 / mag + 1.0f) * 0.5f;
    }
}

extern "C" void kernel_launch(void* const* d_in, const int* in_sizes, int n_in,
                              void* d_out, int out_size, void* d_ws, size_t ws_size,
                              hipStream_t stream) {
    const float* X = (const float*)d_in[0];   // [B, 1024] fp32
    const float* W = (const float*)d_in[1];   // [6*1024] fp32
    float* out = (float*)d_out;               // [B] fp32
    const int B = in_sizes[0] / TN;           // 4096

    const size_t smem_bytes =
        (size_t)(2 * 1024) * sizeof(float) +                      // Dre/Dim
        (size_t)512 * sizeof(float4) +                            // packed twiddle LUT
        (size_t)(2 * LN * TN) * sizeof(float) +                   // exp(iw)/sqrt2 planes
        (size_t)(2 * SPW * TN + 2 * SPW * 1024) * sizeof(float);  // z + B planes

    (void)hipFuncSetAttribute((const void*)phasor_disc_kernel,
                              hipFuncAttributeMaxDynamicSharedMemorySize,
                              (int)smem_bytes);

    dim3 grid(B / SPW);   // 512 workgroups x 256 threads (8 waves, 1 sample each)
    phasor_disc_kernel<<<grid, THREADS, smem_bytes, stream>>>(X, W, out);
}